// Sims_12275016532366
// MI455X (gfx1250) — compile-verified
//
#include <hip/hip_runtime.h>

typedef float v2f __attribute__((ext_vector_type(2)));
typedef float v8f __attribute__((ext_vector_type(8)));

#define NB    16
#define SLEN  1024
#define ELEN  512
#define TEMPC 13.544f
#define EPSC  1e-5f

// d_ws layout:
//   [0..15]   : double sum, double sumsq
//   [16..]    : float sq[NB*SLEN]  (64 KB)

__global__ void init_acc_kernel(double* acc) {
    if (threadIdx.x == 0) { acc[0] = 0.0; acc[1] = 0.0; }
}

// One wave (32 lanes) per row of x: sq[row] = sum(x[row,:]^2)
__global__ void __launch_bounds__(256)
sq_kernel(const float* __restrict__ x, float* __restrict__ sq) {
    int wave = threadIdx.x >> 5;
    int lane = threadIdx.x & 31;
    int row  = blockIdx.x * 8 + wave;            // 0 .. NB*SLEN-1
    const float* xp = x + (size_t)row * ELEN;
    float s = 0.f;
    #pragma unroll
    for (int i = 0; i < ELEN / 32; ++i) {
        float v = xp[lane + i * 32];
        s += v * v;
    }
    #pragma unroll
    for (int off = 16; off > 0; off >>= 1)
        s += __shfl_down(s, off, 32);
    if (lane == 0) sq[row] = s;
}

// Each wave computes a 16x64 strip of the per-batch Gram matrix with
// V_WMMA_F32_16X16X4_F32, converts to relu'd squared distance, writes to d,
// and accumulates global sum / sum-of-squares.
__global__ void __launch_bounds__(256)
gram_dist_kernel(const float* __restrict__ x, const float* __restrict__ sq,
                 float* __restrict__ d, double* __restrict__ acc) {
    int wave = threadIdx.x >> 5;
    int lane = threadIdx.x & 31;
    int gid  = blockIdx.x * 8 + wave;            // 0 .. 16383
    int b    = gid >> 10;                        // batch
    int rem  = gid & 1023;
    int s0   = (rem >> 4) << 4;                  // 16-row tile in s
    int t0   = (rem & 15) << 6;                  // 64-col strip in t

    int half = lane >> 4;                        // 0 or 1
    int lm   = lane & 15;

    const float* xb   = x + (size_t)b * SLEN * ELEN;
    const float* arow = xb + (size_t)(s0 + lm) * ELEN + 2 * half;
    const float* brow0 = xb + (size_t)(t0 +  0 + lm) * ELEN + 2 * half;
    const float* brow1 = xb + (size_t)(t0 + 16 + lm) * ELEN + 2 * half;
    const float* brow2 = xb + (size_t)(t0 + 32 + lm) * ELEN + 2 * half;
    const float* brow3 = xb + (size_t)(t0 + 48 + lm) * ELEN + 2 * half;

    v8f c0 = {}, c1 = {}, c2 = {}, c3 = {};
    #pragma unroll 4
    for (int k0 = 0; k0 < ELEN; k0 += 4) {
        // ISA 32-bit A 16x4 layout: VGPR0 = K {0|2}, VGPR1 = K {1|3} per lane half.
        v2f a  = *(const v2f*)(arow  + k0);
        v2f b0 = *(const v2f*)(brow0 + k0);
        v2f b1 = *(const v2f*)(brow1 + k0);
        v2f b2 = *(const v2f*)(brow2 + k0);
        v2f b3 = *(const v2f*)(brow3 + k0);
        c0 = __builtin_amdgcn_wmma_f32_16x16x4_f32(false, a, false, b0, (short)0, c0, false, false);
        c1 = __builtin_amdgcn_wmma_f32_16x16x4_f32(false, a, false, b1, (short)0, c1, false, false);
        c2 = __builtin_amdgcn_wmma_f32_16x16x4_f32(false, a, false, b2, (short)0, c2, false, false);
        c3 = __builtin_amdgcn_wmma_f32_16x16x4_f32(false, a, false, b3, (short)0, c3, false, false);
    }

    // Epilogue: d = max(0, sq_s + sq_t - 2*gram)
    const float* sqb = sq + b * SLEN;
    float sqt0 = sqb[t0 +  0 + lm];
    float sqt1 = sqb[t0 + 16 + lm];
    float sqt2 = sqb[t0 + 32 + lm];
    float sqt3 = sqb[t0 + 48 + lm];

    double psum = 0.0, psq = 0.0;
    #pragma unroll
    for (int j = 0; j < 8; ++j) {
        int s_row = s0 + j + 8 * half;           // C/D layout: VGPR j -> M = j | j+8
        float sqs = sqb[s_row];
        size_t rowbase = ((size_t)b * SLEN + s_row) * SLEN;
        float v0 = fmaxf(0.f, sqs + sqt0 - 2.f * c0[j]);
        float v1 = fmaxf(0.f, sqs + sqt1 - 2.f * c1[j]);
        float v2 = fmaxf(0.f, sqs + sqt2 - 2.f * c2[j]);
        float v3 = fmaxf(0.f, sqs + sqt3 - 2.f * c3[j]);
        d[rowbase + t0 +  0 + lm] = v0;
        d[rowbase + t0 + 16 + lm] = v1;
        d[rowbase + t0 + 32 + lm] = v2;
        d[rowbase + t0 + 48 + lm] = v3;
        psum += (double)v0 + (double)v1 + (double)v2 + (double)v3;
        psq  += (double)v0 * v0 + (double)v1 * v1 + (double)v2 * v2 + (double)v3 * v3;
    }

    // Block reduction then two f64 atomics (global_atomic_add_f64).
    __shared__ double red0[256];
    __shared__ double red1[256];
    int tid = threadIdx.x;
    red0[tid] = psum;
    red1[tid] = psq;
    __syncthreads();
    #pragma unroll
    for (int s = 128; s > 0; s >>= 1) {
        if (tid < s) { red0[tid] += red0[tid + s]; red1[tid] += red1[tid + s]; }
        __syncthreads();
    }
    if (tid == 0) {
        atomicAdd(&acc[0], red0[0]);
        atomicAdd(&acc[1], red1[0]);
    }
}

// One block per output row: normalize with global mean/var, then softmax(-out/T).
__global__ void __launch_bounds__(256)
softmax_kernel(float* __restrict__ d, const double* __restrict__ acc,
               const float* __restrict__ gamma, const float* __restrict__ beta) {
    const double count = (double)NB * SLEN * SLEN;
    double mean = acc[0] / count;
    double var  = acc[1] / count - mean * mean;
    float  fmean = (float)mean;
    float  inv   = rsqrtf((float)var + EPSC);
    float  g  = gamma[0];
    float  be = beta[0];

    size_t base = (size_t)blockIdx.x * SLEN;
    int tid = threadIdx.x;

    float vals[4];
    float mymax = -3.0e38f;
    #pragma unroll
    for (int i = 0; i < 4; ++i) {
        float dv = d[base + tid + i * 256];
        float t  = -(((dv - fmean) * inv) * g + be) / TEMPC;
        vals[i]  = t;
        mymax    = fmaxf(mymax, t);
    }

    __shared__ float red[256];
    red[tid] = mymax;
    __syncthreads();
    #pragma unroll
    for (int s = 128; s > 0; s >>= 1) {
        if (tid < s) red[tid] = fmaxf(red[tid], red[tid + s]);
        __syncthreads();
    }
    float rmax = red[0];
    __syncthreads();

    float psum = 0.f;
    #pragma unroll
    for (int i = 0; i < 4; ++i) {
        vals[i] = __expf(vals[i] - rmax);
        psum += vals[i];
    }
    red[tid] = psum;
    __syncthreads();
    #pragma unroll
    for (int s = 128; s > 0; s >>= 1) {
        if (tid < s) red[tid] += red[tid + s];
        __syncthreads();
    }
    float rinv = 1.f / red[0];
    #pragma unroll
    for (int i = 0; i < 4; ++i)
        d[base + tid + i * 256] = vals[i] * rinv;
}

extern "C" void kernel_launch(void* const* d_in, const int* in_sizes, int n_in,
                              void* d_out, int out_size, void* d_ws, size_t ws_size,
                              hipStream_t stream) {
    const float* x     = (const float*)d_in[0];
    const float* gamma = (const float*)d_in[1];
    const float* beta  = (const float*)d_in[2];
    float*  out = (float*)d_out;
    double* acc = (double*)d_ws;
    float*  sq  = (float*)((char*)d_ws + 16);

    init_acc_kernel<<<1, 32, 0, stream>>>(acc);
    sq_kernel<<<(NB * SLEN) / 8, 256, 0, stream>>>(x, sq);
    gram_dist_kernel<<<(NB * (SLEN / 16) * (SLEN / 64)) / 8, 256, 0, stream>>>(x, sq, out, acc);
    softmax_kernel<<<NB * SLEN, 256, 0, stream>>>(out, acc, gamma, beta);
}